// EnhancedClassifier_27169963114789
// MI455X (gfx1250) — compile-verified
//
#include <hip/hip_runtime.h>

typedef __attribute__((ext_vector_type(2))) float v2f;
typedef __attribute__((ext_vector_type(8))) float v8f;

// ---------------- utility kernels ----------------

__global__ void k_fill0(float* __restrict__ p, int n) {
    int i = blockIdx.x * blockDim.x + threadIdx.x;
    if (i < n) p[i] = 0.f;
}

__global__ void k_deg(const long long* __restrict__ dst, float* __restrict__ deg, int E) {
    int e = blockIdx.x * blockDim.x + threadIdx.x;
    if (e < E) atomicAdd(&deg[dst[e]], 1.0f);
}

__global__ void k_deginv(float* __restrict__ deg, int n) {
    int i = blockIdx.x * blockDim.x + threadIdx.x;
    if (i < n) deg[i] = 1.0f / fmaxf(deg[i], 1.0f);
}

// scatter-add h[src] rows into agg[dst] rows; F features, 1<<shift float4-chunks per edge
__global__ void k_scatter(const float* __restrict__ h, const long long* __restrict__ src,
                          const long long* __restrict__ dst, float* __restrict__ agg,
                          int E, int F, int shift) {
    long long idx = (long long)blockIdx.x * blockDim.x + threadIdx.x;
    int e = (int)(idx >> shift);
    if (e >= E) return;
    int c = (int)(idx & ((1 << shift) - 1));
    long long s = src[e], d = dst[e];
    const float4 v = *(const float4*)(h + (size_t)s * F + 4 * c);
    float* o = agg + (size_t)d * F + 4 * c;
    atomicAdd(o + 0, v.x);
    atomicAdd(o + 1, v.y);
    atomicAdd(o + 2, v.z);
    atomicAdd(o + 3, v.w);
}

__global__ void k_scatter1(const float* __restrict__ sv, const long long* __restrict__ src,
                           const long long* __restrict__ dst, float* __restrict__ acc, int E) {
    int e = blockIdx.x * blockDim.x + threadIdx.x;
    if (e < E) atomicAdd(&acc[dst[e]], sv[src[e]]);
}

__global__ void k_final(const float* __restrict__ acc, const float* __restrict__ deginv,
                        const float* __restrict__ rv, float* __restrict__ out, int n) {
    int i = blockIdx.x * blockDim.x + threadIdx.x;
    if (i < n) out[i] = acc[i] * deginv[i] + rv[i];
}

// ---------------- WMMA GEMM: out[64 rows x 128 cols per block] ----------------
// out = (scale1 ? diag(scale1)*A1 : A1) @ W1 [+ A2 @ W2] + bias, optional ReLU.
// 256 threads = 8 wave32; wave w owns column tile [16w, 16w+16) across 4 M-tiles.
// Each B fragment is loaded once per k-step and reused by 4 WMMAs (4 accumulators).
template <int K>
__global__ __launch_bounds__(256) void k_gemm(
    const float* __restrict__ A1, const float* __restrict__ scale1,
    const float* __restrict__ W1,
    const float* __restrict__ A2, const float* __restrict__ W2,
    const float* __restrict__ bias, float* __restrict__ out,
    int relu, int nrows) {
    __shared__ float lA[64 * K];

    const int tid = threadIdx.x;
    const int row0 = blockIdx.x * 64;
    const int wave = tid >> 5;
    const int lane = tid & 31;
    const int m = lane & 15;               // M within 16-tile / N within column tile
    const int koff = (lane >> 4) << 1;     // upper half-wave holds K+2, K+3
    const int n0 = wave * 16;

    v8f c0 = {0.f, 0.f, 0.f, 0.f, 0.f, 0.f, 0.f, 0.f};
    v8f c1 = c0, c2 = c0, c3 = c0;

    // ---- pass 1: A1 (optionally row-scaled) @ W1 ----
    for (int i = tid; i < 64 * K; i += 256) {
        int r = i / K, k = i % K;
        int row = row0 + r;
        float v = 0.f;
        if (row < nrows) {
            v = A1[(size_t)row * K + k];
            if (scale1) v *= scale1[row];
        }
        lA[i] = v;
    }
    __syncthreads();
    #pragma unroll 4
    for (int k = 0; k < K; k += 4) {
        v2f b;
        b.x = W1[(size_t)(k + koff) * 128 + n0 + m];
        b.y = W1[(size_t)(k + koff + 1) * 128 + n0 + m];
        v2f a0 = *(const v2f*)&lA[(0 * 16 + m) * K + k + koff];
        v2f a1 = *(const v2f*)&lA[(1 * 16 + m) * K + k + koff];
        v2f a2 = *(const v2f*)&lA[(2 * 16 + m) * K + k + koff];
        v2f a3 = *(const v2f*)&lA[(3 * 16 + m) * K + k + koff];
        c0 = __builtin_amdgcn_wmma_f32_16x16x4_f32(false, a0, false, b, (short)0, c0, false, false);
        c1 = __builtin_amdgcn_wmma_f32_16x16x4_f32(false, a1, false, b, (short)0, c1, false, false);
        c2 = __builtin_amdgcn_wmma_f32_16x16x4_f32(false, a2, false, b, (short)0, c2, false, false);
        c3 = __builtin_amdgcn_wmma_f32_16x16x4_f32(false, a3, false, b, (short)0, c3, false, false);
    }

    // ---- pass 2 (optional): + A2 @ W2 ----
    if (A2) {
        __syncthreads();
        for (int i = tid; i < 64 * K; i += 256) {
            int r = i / K, k = i % K;
            int row = row0 + r;
            lA[i] = (row < nrows) ? A2[(size_t)row * K + k] : 0.f;
        }
        __syncthreads();
        #pragma unroll 4
        for (int k = 0; k < K; k += 4) {
            v2f b;
            b.x = W2[(size_t)(k + koff) * 128 + n0 + m];
            b.y = W2[(size_t)(k + koff + 1) * 128 + n0 + m];
            v2f a0 = *(const v2f*)&lA[(0 * 16 + m) * K + k + koff];
            v2f a1 = *(const v2f*)&lA[(1 * 16 + m) * K + k + koff];
            v2f a2 = *(const v2f*)&lA[(2 * 16 + m) * K + k + koff];
            v2f a3 = *(const v2f*)&lA[(3 * 16 + m) * K + k + koff];
            c0 = __builtin_amdgcn_wmma_f32_16x16x4_f32(false, a0, false, b, (short)0, c0, false, false);
            c1 = __builtin_amdgcn_wmma_f32_16x16x4_f32(false, a1, false, b, (short)0, c1, false, false);
            c2 = __builtin_amdgcn_wmma_f32_16x16x4_f32(false, a2, false, b, (short)0, c2, false, false);
            c3 = __builtin_amdgcn_wmma_f32_16x16x4_f32(false, a3, false, b, (short)0, c3, false, false);
        }
    }

    // ---- store: C layout — VGPR r: lanes 0-15 -> M=r, lanes 16-31 -> M=r+8 ----
    float bv = bias[n0 + m];
    const int mhalf = (lane >> 4) << 3;
    #pragma unroll
    for (int t = 0; t < 4; t++) {
        const v8f& c = (t == 0) ? c0 : (t == 1) ? c1 : (t == 2) ? c2 : c3;
        const int mbase = row0 + t * 16 + mhalf;
        #pragma unroll
        for (int r = 0; r < 8; r++) {
            int row = mbase + r;
            if (row < nrows) {
                float v = c[r] + bv;
                if (relu) v = fmaxf(v, 0.f);
                out[(size_t)row * 128 + n0 + m] = v;
            }
        }
    }
}

// ---------------- LayerNorm + ReLU + residual-add (one wave32 per 128-wide row) ----------------
__global__ __launch_bounds__(256) void k_ln_relu_add(
    const float* __restrict__ t, const float* __restrict__ g, const float* __restrict__ b,
    float* __restrict__ res /* in: x@Wres+bres, out: h1 */, int nrows) {
    int row = blockIdx.x * 8 + (threadIdx.x >> 5);
    int lane = threadIdx.x & 31;
    if (row >= nrows) return;
    float4 v = *(const float4*)(t + (size_t)row * 128 + lane * 4);
    float s = v.x + v.y + v.z + v.w;
    for (int o = 16; o > 0; o >>= 1) s += __shfl_xor(s, o, 32);
    float mu = s * (1.f / 128.f);
    float dx = v.x - mu, dy = v.y - mu, dz = v.z - mu, dw = v.w - mu;
    float q = dx * dx + dy * dy + dz * dz + dw * dw;
    for (int o = 16; o > 0; o >>= 1) q += __shfl_xor(q, o, 32);
    float rs = rsqrtf(q * (1.f / 128.f) + 1e-5f);
    float* rr = res + (size_t)row * 128 + lane * 4;
    float4 rvv = *(const float4*)rr;
    int f = lane * 4;
    float4 o4;
    o4.x = fmaxf(dx * rs * g[f + 0] + b[f + 0], 0.f) + rvv.x;
    o4.y = fmaxf(dy * rs * g[f + 1] + b[f + 1], 0.f) + rvv.y;
    o4.z = fmaxf(dz * rs * g[f + 2] + b[f + 2], 0.f) + rvv.z;
    o4.w = fmaxf(dw * rs * g[f + 3] + b[f + 3], 0.f) + rvv.w;
    *(float4*)rr = o4;
}

// ---------------- layer-4 head GEMV: s = h@Wl4 ; r = h@Wr4 + b4 ----------------
__global__ __launch_bounds__(256) void k_head(
    const float* __restrict__ h, const float* __restrict__ wl, const float* __restrict__ wr,
    const float* __restrict__ b4, float* __restrict__ sv, float* __restrict__ rv, int nrows) {
    int row = blockIdx.x * 8 + (threadIdx.x >> 5);
    int lane = threadIdx.x & 31;
    if (row >= nrows) return;
    const float4 v = *(const float4*)(h + (size_t)row * 128 + lane * 4);
    const float4 a = *(const float4*)(wl + lane * 4);
    const float4 c = *(const float4*)(wr + lane * 4);
    float sd = v.x * a.x + v.y * a.y + v.z * a.z + v.w * a.w;
    float rd = v.x * c.x + v.y * c.y + v.z * c.z + v.w * c.w;
    for (int o = 16; o > 0; o >>= 1) {
        sd += __shfl_xor(sd, o, 32);
        rd += __shfl_xor(rd, o, 32);
    }
    if (lane == 0) {
        sv[row] = sd;
        rv[row] = rd + b4[0];
    }
}

// ---------------- launcher ----------------
extern "C" void kernel_launch(void* const* d_in, const int* in_sizes, int n_in,
                              void* d_out, int out_size, void* d_ws, size_t ws_size,
                              hipStream_t stream) {
    const float* x    = (const float*)d_in[0];
    const long long* ei = (const long long*)d_in[1];
    const float* Wl1  = (const float*)d_in[2];
    const float* Wr1  = (const float*)d_in[3];
    const float* b1   = (const float*)d_in[4];
    const float* ln_g = (const float*)d_in[5];
    const float* ln_b = (const float*)d_in[6];
    const float* Wres = (const float*)d_in[7];
    const float* bres = (const float*)d_in[8];
    const float* Wl2  = (const float*)d_in[9];
    const float* Wr2  = (const float*)d_in[10];
    const float* b2   = (const float*)d_in[11];
    const float* Wl3  = (const float*)d_in[12];
    const float* Wr3  = (const float*)d_in[13];
    const float* b3   = (const float*)d_in[14];
    const float* Wl4  = (const float*)d_in[15];
    const float* Wr4  = (const float*)d_in[16];
    const float* b4   = (const float*)d_in[17];

    const int N = in_sizes[0] / 64;
    const int E = in_sizes[1] / 2;
    const long long* src = ei;
    const long long* dst = ei + E;

    float* ws  = (float*)d_ws;
    float* deg = ws;                        // N floats (becomes deg_inv in place)
    float* agg = ws + N;                    // N*128
    float* hA  = agg + (size_t)N * 128;     // N*128
    float* hB  = hA + (size_t)N * 128;      // N*128
    float* sv  = hB + (size_t)N * 128;      // N
    float* rv  = sv + N;                    // N
    float* oac = rv + N;                    // N

    const int B = 256;
    auto blk = [](long long n) { return (int)((n + 255) / 256); };
    const int gblocks = (N + 63) / 64;

    // degrees -> deg_inv
    k_fill0<<<blk(N), B, 0, stream>>>(deg, N);
    k_deg<<<blk(E), B, 0, stream>>>(dst, deg, E);
    k_deginv<<<blk(N), B, 0, stream>>>(deg, N);

    // ---- layer 1: aggregate 64-wide, dual GEMM, residual GEMM, LN+ReLU+add ----
    k_fill0<<<blk((long long)N * 64), B, 0, stream>>>(agg, N * 64);
    k_scatter<<<blk((long long)E * 16), B, 0, stream>>>(x, src, dst, agg, E, 64, 4);
    k_gemm<64><<<gblocks, 256, 0, stream>>>(agg, deg, Wl1, x, Wr1, b1, hA, 0, N);
    k_gemm<64><<<gblocks, 256, 0, stream>>>(x, nullptr, Wres, nullptr, nullptr, bres, hB, 0, N);
    k_ln_relu_add<<<(N + 7) / 8, 256, 0, stream>>>(hA, ln_g, ln_b, hB, N);   // h1 -> hB

    // ---- layer 2 ----
    k_fill0<<<blk((long long)N * 128), B, 0, stream>>>(agg, N * 128);
    k_scatter<<<blk((long long)E * 32), B, 0, stream>>>(hB, src, dst, agg, E, 128, 5);
    k_gemm<128><<<gblocks, 256, 0, stream>>>(agg, deg, Wl2, hB, Wr2, b2, hA, 1, N); // h2 -> hA

    // ---- layer 3 ----
    k_fill0<<<blk((long long)N * 128), B, 0, stream>>>(agg, N * 128);
    k_scatter<<<blk((long long)E * 32), B, 0, stream>>>(hA, src, dst, agg, E, 128, 5);
    k_gemm<128><<<gblocks, 256, 0, stream>>>(agg, deg, Wl3, hA, Wr3, b3, hB, 1, N); // h3 -> hB

    // ---- layer 4: transform-then-aggregate (mean commutes with linear map) ----
    k_head<<<(N + 7) / 8, 256, 0, stream>>>(hB, Wl4, Wr4, b4, sv, rv, N);
    k_fill0<<<blk(N), B, 0, stream>>>(oac, N);
    k_scatter1<<<blk(E), B, 0, stream>>>(sv, src, dst, oac, E);
    k_final<<<blk(N), B, 0, stream>>>(oac, deg, rv, (float*)d_out, N);
}